// SRHT_85710367359547
// MI455X (gfx1250) — compile-verified
//
#include <hip/hip_runtime.h>

// ---------------------------------------------------------------------------
// SRHT on gfx1250 (MI455X): FWHT_1024 = H32(hi5) (x) H32(lo5), each H32 done
// as VALU half-combine + H16 matmul on the f32 WMMA pipe (16x16x4, K-chunked).
// Memory-bound (~320MB @ 23.3TB/s); one global read of x, one NT write of out.
// ---------------------------------------------------------------------------

typedef float v2f __attribute__((ext_vector_type(2)));
typedef float v8f __attribute__((ext_vector_type(8)));

#define D_IN    1024
#define NBLK    4
#define TROWS   16      // batch rows per workgroup tile (WMMA N dimension)
#define THREADS 256     // 8 wave32 waves

// XOR-swizzled LDS word address for element i (0..1023) of row n (0..15).
// Keeps both the stride-1 and stride-32 gathers at
// <=2-way bank conflicts on the 64-bank LDS.
__device__ __forceinline__ int lw(int i, int n) {
    return (i << 4) + (n ^ (i & 15));
}

// One H32 pass along one 5-bit axis of the 1024-vector, for all 16 rows.
// STRIDE_SEL=0: axis is low 5 bits (element = c*32 + k)
// STRIDE_SEL=1: axis is high 5 bits (element = k*32 + c)
// H32 = H2 (x) H16:  out[0:16] = H16*(lo+hi), out[16:32] = H16*(lo-hi).
// A = H16 (symmetric, entries +-scaleA), built analytically per lane.
// RAD: multiply inputs by radp[element] (diagonal re-randomization fold).
// Combos c are disjoint element sets across waves -> no barrier inside pass.
template <int STRIDE_SEL, bool RAD>
__device__ __forceinline__ void h32_pass(float* __restrict__ tile,
                                         const float* __restrict__ radp,
                                         float scaleA, int lane, int wave) {
    const int n = lane & 15;   // B/C/D column (batch row)
    const int m = lane & 15;   // A row
    const int h = lane >> 4;   // half-wave select
#pragma unroll
    for (int q = 0; q < 4; ++q) {
        const int c = wave + (q << 3);   // 0..31, disjoint per wave
        v8f acc_s = {0.f, 0.f, 0.f, 0.f, 0.f, 0.f, 0.f, 0.f};
        v8f acc_d = {0.f, 0.f, 0.f, 0.f, 0.f, 0.f, 0.f, 0.f};
#pragma unroll
        for (int k0 = 0; k0 < 16; k0 += 4) {
            // K-chunk lane mapping (ISA 7.12.2): vgpr0 holds K=k0+2h, vgpr1 K=k0+2h+1
            const int ka = k0 + 2 * h;
            const int kb = ka + 1;
            const int ia_lo = STRIDE_SEL ? (ka * 32 + c)        : (c * 32 + ka);
            const int ib_lo = STRIDE_SEL ? (kb * 32 + c)        : (c * 32 + kb);
            const int ia_hi = STRIDE_SEL ? ((ka + 16) * 32 + c) : (c * 32 + ka + 16);
            const int ib_hi = STRIDE_SEL ? ((kb + 16) * 32 + c) : (c * 32 + kb + 16);
            float xa_lo = tile[lw(ia_lo, n)];
            float xb_lo = tile[lw(ib_lo, n)];
            float xa_hi = tile[lw(ia_hi, n)];
            float xb_hi = tile[lw(ib_hi, n)];
            if (RAD) {   // broadcast-address global loads, L0-resident (4KB)
                xa_lo *= radp[ia_lo];
                xb_lo *= radp[ib_lo];
                xa_hi *= radp[ia_hi];
                xb_hi *= radp[ib_hi];
            }
            v2f bs, bd, a;
            bs.x = xa_lo + xa_hi;  bs.y = xb_lo + xb_hi;   // H2 sum half
            bd.x = xa_lo - xa_hi;  bd.y = xb_lo - xb_hi;   // H2 diff half
            a.x = (__popc(m & ka) & 1) ? -scaleA : scaleA; // H16[m][ka]
            a.y = (__popc(m & kb) & 1) ? -scaleA : scaleA; // H16[m][kb]
            acc_s = __builtin_amdgcn_wmma_f32_16x16x4_f32(
                false, a, false, bs, (short)0, acc_s, false, false);
            acc_d = __builtin_amdgcn_wmma_f32_16x16x4_f32(
                false, a, false, bd, (short)0, acc_d, false, false);
        }
        // D layout: vgpr r holds M = r + 8*h, columns = lanes (mod 16)
#pragma unroll
        for (int r = 0; r < 8; ++r) {
            const int mo  = r + 8 * h;
            const int i_s = STRIDE_SEL ? (mo * 32 + c)        : (c * 32 + mo);
            const int i_d = STRIDE_SEL ? ((mo + 16) * 32 + c) : (c * 32 + mo + 16);
            tile[lw(i_s, n)] = acc_s[r];
            tile[lw(i_d, n)] = acc_d[r];
        }
    }
}

__global__ __launch_bounds__(THREADS) void srht_wmma_kernel(
    const float* __restrict__ x,     // (B, 1024)
    const float* __restrict__ rad,   // (3, 4, 1024)
    const int*   __restrict__ perm,  // (nfeat,)
    float*       __restrict__ out,   // (B, nfeat)
    int nfeat) {
    __shared__ float tile[TROWS * D_IN];   // 64 KB, xor-swizzled

    const int wg   = blockIdx.x;
    const int nb   = wg & (NBLK - 1);         // block index 0..3
    const int r0   = (wg >> 2) * TROWS;       // first batch row of tile
    const int tid  = threadIdx.x;
    const int lane = tid & 31;
    const int wave = tid >> 5;

    const float* rad0 = rad + (size_t)(0 * NBLK + nb) * D_IN;
    const float* rad1 = rad + (size_t)(1 * NBLK + nb) * D_IN;
    const float* rad2 = rad + (size_t)(2 * NBLK + nb) * D_IN;

    // ---- fill LDS: tile = x * rad0 (coalesced float4 reads) ----
#pragma unroll
    for (int it = 0; it < (TROWS * D_IN / 4) / THREADS; ++it) {
        const int e4 = tid + THREADS * it;    // 0..4095
        const int n  = e4 >> 8;               // row in tile
        const int i  = (e4 & 255) << 2;       // element (multiple of 4)
        const float4 v  = *(const float4*)(x + (size_t)(r0 + n) * D_IN + i);
        const float4 rv = *(const float4*)(rad0 + i);
        tile[lw(i + 0, n)] = v.x * rv.x;
        tile[lw(i + 1, n)] = v.y * rv.y;
        tile[lw(i + 2, n)] = v.z * rv.z;
        tile[lw(i + 3, n)] = v.w * rv.w;
    }
    __syncthreads();

    const float inv32 = 0.03125f;   // 1/sqrt(1024), folded into A (exact)

    // FWHT #1
    h32_pass<0, false>(tile, nullptr, 1.0f, lane, wave);  __syncthreads();
    h32_pass<1, false>(tile, nullptr, 1.0f, lane, wave);  __syncthreads();
    // FWHT #2 (inputs scaled by rad1/32, folded into load+A)
    h32_pass<0, true >(tile, rad1,   inv32, lane, wave);  __syncthreads();
    h32_pass<1, false>(tile, nullptr, 1.0f, lane, wave);  __syncthreads();
    // FWHT #3 (inputs scaled by rad2/32)
    h32_pass<0, true >(tile, rad2,   inv32, lane, wave);  __syncthreads();
    h32_pass<1, false>(tile, nullptr, 1.0f, lane, wave);  __syncthreads();

    // ---- permutation gather -> coalesced NT stores of this block's columns
    for (int j = tid; j < nfeat; j += THREADS) {
        const int p = perm[j];
        if ((p >> 10) == nb) {                 // column belongs to our block
            const int pi = p & (D_IN - 1);
#pragma unroll
            for (int n = 0; n < TROWS; ++n) {
                const float v = tile[lw(pi, n)];
                __builtin_nontemporal_store(
                    v, out + (size_t)(r0 + n) * nfeat + j);
            }
        }
    }
}

extern "C" void kernel_launch(void* const* d_in, const int* in_sizes, int n_in,
                              void* d_out, int out_size, void* d_ws, size_t ws_size,
                              hipStream_t stream) {
    const float* x    = (const float*)d_in[0];
    const float* rad  = (const float*)d_in[1];
    const int*   perm = (const int*)d_in[2];
    float*       out  = (float*)d_out;

    const int B      = in_sizes[0] / D_IN;   // 16384 (multiple of TROWS)
    const int nfeat  = in_sizes[2];          // 4096
    const int ntiles = B / TROWS;            // 1024
    dim3 grid(ntiles * NBLK);                // one WG per (row-tile, block)
    dim3 block(THREADS);
    srht_wmma_kernel<<<grid, block, 0, stream>>>(x, rad, perm, out, nfeat);
}